// PenalizeLoss_86629490360770
// MI455X (gfx1250) — compile-verified
//
#include <hip/hip_runtime.h>
#include <math.h>
#include <stdint.h>

// Problem constants (match reference: B=1048576, C=128, WEIGHT=0.5)
#define NCLS 128
#define BLOCKS 2048
#define WAVES_PER_BLOCK 8
#define ROWS_PER_WAVE 64   // 2048 * 8 * 64 = 1,048,576 rows
#define NBUF 4             // async pipeline: 4 LDS buffers, 3 outstanding prefetches
#define PENALTY_WEIGHT 0.5

// ---- wave32 reductions (warpSize == 32 on gfx1250) ----
__device__ __forceinline__ float wave_sum32(float v) {
    for (int o = 16; o > 0; o >>= 1) v += __shfl_xor(v, o, 32);
    return v;
}
// argmax with first-occurrence tie-break (matches jnp.argmax); also yields the max value
__device__ __forceinline__ void wave_argmax32(float& v, int& i) {
    for (int o = 16; o > 0; o >>= 1) {
        float v2 = __shfl_xor(v, o, 32);
        int   i2 = __shfl_xor(i, o, 32);
        if (v2 > v || (v2 == v && i2 < i)) { v = v2; i = i2; }
    }
}

// One row (512B) as a per-lane 16B async copy global -> LDS (ASYNCcnt path).
// Streaming data: non-temporal hint keeps the 512MB stream from thrashing L2.
__device__ __forceinline__ void async_row_to_lds(uint32_t lds_off, const float* gptr) {
    uint64_t ga = (uint64_t)(uintptr_t)gptr;
    asm volatile("global_load_async_to_lds_b128 %0, %1, off th:TH_LOAD_NT"
                 :: "v"(lds_off), "v"(ga) : "memory");
}

__global__ __launch_bounds__(256) void pl_main(const float* __restrict__ P,
                                               const int*   __restrict__ T,
                                               float* __restrict__ ce_part,
                                               float* __restrict__ pen_part,
                                               int*   __restrict__ cnt_part) {
    __shared__ __align__(16) float stage[WAVES_PER_BLOCK][NBUF][NCLS]; // 16 KB ring
    __shared__ float s_ce[WAVES_PER_BLOCK];
    __shared__ float s_pen[WAVES_PER_BLOCK];
    __shared__ int   s_cnt[WAVES_PER_BLOCK];

    const int lane = threadIdx.x & 31;
    const int wave = threadIdx.x >> 5;
    const long long wave_gid = (long long)blockIdx.x * WAVES_PER_BLOCK + wave;
    const long long row0 = wave_gid * ROWS_PER_WAVE;

    // addrspace(3) byte offset of this wave's ring buffer (per-lane 16B slot)
    const uint32_t lds_base = (uint32_t)(uintptr_t)&stage[wave][0][lane * 4];
    const float* gbase = P + row0 * NCLS + lane * 4;

    // preload all 64 targets for this wave (2 coalesced loads), broadcast later via shfl
    const int tpack0 = T[row0 + lane];
    const int tpack1 = T[row0 + 32 + lane];

    // prime the pipeline: rows 0..2 -> buffers 0..2
    async_row_to_lds(lds_base + 0 * (NCLS * 4), gbase + 0 * NCLS);
    async_row_to_lds(lds_base + 1 * (NCLS * 4), gbase + 1 * NCLS);
    async_row_to_lds(lds_base + 2 * (NCLS * 4), gbase + 2 * NCLS);

    float ce_acc = 0.0f, pen_acc = 0.0f;
    int cnt_acc = 0;

    for (int i = 0; i < ROWS_PER_WAVE; ++i) {
        const int st = i & (NBUF - 1);

        // keep 3 prefetches in flight; ASYNCcnt completes in order, so a wait
        // of (outstanding-1) guarantees the oldest (row i) has landed.
        if (i + 3 < ROWS_PER_WAVE) {
            // buffer (i+3)&3 was last read at iteration i-1 and dscnt-drained there
            async_row_to_lds(lds_base + ((i + 3) & (NBUF - 1)) * (NCLS * 4),
                             gbase + (long long)(i + 3) * NCLS);
            asm volatile("s_wait_asynccnt 3" ::: "memory");
        } else if (i + 2 < ROWS_PER_WAVE) {
            asm volatile("s_wait_asynccnt 2" ::: "memory");
        } else if (i + 1 < ROWS_PER_WAVE) {
            asm volatile("s_wait_asynccnt 1" ::: "memory");
        } else {
            asm volatile("s_wait_asynccnt 0" ::: "memory");
        }

        // target for this row: register broadcast instead of a uniform vmem load
        const int t = __shfl((i < 32) ? tpack0 : tpack1, i & 31, 32);

        // all LDS reads of this buffer happen here, then drain DScnt so a later
        // iteration may async-overwrite it safely
        const float4 x4 = *(const float4*)&stage[wave][st][lane * 4];
        const float  xt = stage[wave][st][t];  // broadcast read
        const float  x0 = stage[wave][st][0];  // broadcast read
        asm volatile("s_wait_dscnt 0" ::: "memory");

        // fused max+argmax (first index wins on ties); bv ends as the row max
        int   bi = lane * 4;
        float bv = x4.x;
        if (x4.y > bv) { bv = x4.y; bi = lane * 4 + 1; }
        if (x4.z > bv) { bv = x4.z; bi = lane * 4 + 2; }
        if (x4.w > bv) { bv = x4.w; bi = lane * 4 + 3; }
        wave_argmax32(bv, bi);
        const float m = bv;

        // softmax denominator
        float e = expf(x4.x - m) + expf(x4.y - m) + expf(x4.z - m) + expf(x4.w - m);
        float s = wave_sum32(e);

        // CE: -(log_softmax[target]) = m + log(s) - x[target]
        ce_acc += m + logf(s) - xt;

        // penalty rows: target==1 && argmax==0 (wave-uniform branch)
        if ((t == 1) && (bi == 0)) {
            float p0 = expf(x0 - m) / s;
            pen_acc += -log1pf(-p0);
            cnt_acc += 1;
        }
    }

    // per-block deterministic combine (all lanes hold identical accumulators)
    if (lane == 0) { s_ce[wave] = ce_acc; s_pen[wave] = pen_acc; s_cnt[wave] = cnt_acc; }
    __syncthreads();
    if (threadIdx.x == 0) {
        float ce = 0.0f, pen = 0.0f; int c = 0;
        for (int w = 0; w < WAVES_PER_BLOCK; ++w) { ce += s_ce[w]; pen += s_pen[w]; c += s_cnt[w]; }
        ce_part[blockIdx.x]  = ce;
        pen_part[blockIdx.x] = pen;
        cnt_part[blockIdx.x] = c;
    }
}

__global__ __launch_bounds__(256) void pl_final(const float* __restrict__ ce_part,
                                                const float* __restrict__ pen_part,
                                                const int*   __restrict__ cnt_part,
                                                float* __restrict__ out) {
    __shared__ double s1[256];
    __shared__ double s2[256];
    __shared__ long long s3[256];
    const int t = threadIdx.x;
    double ce = 0.0, pen = 0.0; long long cnt = 0;
    for (int i = t; i < BLOCKS; i += 256) {
        ce  += (double)ce_part[i];
        pen += (double)pen_part[i];
        cnt += (long long)cnt_part[i];
    }
    s1[t] = ce; s2[t] = pen; s3[t] = cnt;
    __syncthreads();
    for (int o = 128; o > 0; o >>= 1) {
        if (t < o) { s1[t] += s1[t + o]; s2[t] += s2[t + o]; s3[t] += s3[t + o]; }
        __syncthreads();
    }
    if (t == 0) {
        const double nrows = (double)BLOCKS * WAVES_PER_BLOCK * ROWS_PER_WAVE;
        double ce_loss  = s1[0] / nrows;
        double pen_loss = (s3[0] > 0) ? (s2[0] / (double)s3[0]) : 0.0;
        out[0] = (float)(ce_loss + PENALTY_WEIGHT * pen_loss);
    }
}

extern "C" void kernel_launch(void* const* d_in, const int* in_sizes, int n_in,
                              void* d_out, int out_size, void* d_ws, size_t ws_size,
                              hipStream_t stream) {
    (void)in_sizes; (void)n_in; (void)out_size; (void)ws_size;
    const float* P = (const float*)d_in[0];
    const int*   T = (const int*)d_in[1];

    float* ce_part  = (float*)d_ws;
    float* pen_part = ce_part + BLOCKS;
    int*   cnt_part = (int*)(pen_part + BLOCKS);

    pl_main<<<BLOCKS, 256, 0, stream>>>(P, T, ce_part, pen_part, cnt_part);
    pl_final<<<1, 256, 0, stream>>>(ce_part, pen_part, cnt_part, (float*)d_out);
}